// Gemma4Experts_46969762349450
// MI455X (gfx1250) — compile-verified
//
#include <hip/hip_runtime.h>
#include <stdint.h>

// Problem constants (from reference)
#define T_TOK 8192
#define EXP   8
#define HID   2048
#define INTER 1024
#define TOPK  2

// ---- fused fallback tiling ----
#define TB      16
#define KC      64
#define THREADS 256
#define PADX    (KC + 2)
#define PADA    (INTER + 2)

// ---- two-phase GEMM tiling ----
#define MT      128        // token rows per block
#define NT      128        // output cols per block
#define KCH     32         // K chunk staged in LDS
#define PHPADX  36         // KCH + 4  (keeps 16B alignment for b128 async)
#define PHPADB  132        // NT  + 4  (matches TDM pad: 4 dwords per 128)

// workspace layout
#define OFF_COUNTS 0
#define OFF_OFFS   64
#define OFF_TOK    4096
#define OFF_W      (4096 + (size_t)EXP * T_TOK * 4)
#define OFF_ACT    (1u << 20)
#define WS_NEED    (OFF_ACT + (size_t)T_TOK * TOPK * INTER * 4)

typedef __attribute__((ext_vector_type(2))) float    v2f;
typedef __attribute__((ext_vector_type(8))) float    v8f;
typedef __attribute__((ext_vector_type(4))) uint32_t u32x4;
typedef __attribute__((ext_vector_type(8))) uint32_t u32x8;

__device__ __forceinline__ v8f wmma_f32(v2f a, v2f b, v8f c) {
  // D = A(16x4) * B(4x16) + C(16x16), fp32 — V_WMMA_F32_16X16X4_F32
  return __builtin_amdgcn_wmma_f32_16x16x4_f32(
      false, a, false, b, (short)0, c, false, false);
}

__device__ __forceinline__ float gelu_tanh(float x) {
  float x3 = x * x * x;
  return 0.5f * x * (1.0f + tanhf(0.7978845608028654f * (x + 0.044715f * x3)));
}

// CDNA5 async copy engine: 16B global -> LDS, tracked by ASYNCcnt
__device__ __forceinline__ void async_copy_b128(uint32_t lds_off, const float* gaddr) {
  asm volatile("global_load_async_to_lds_b128 %0, %1, off"
               :: "v"(lds_off), "v"(gaddr) : "memory");
}
__device__ __forceinline__ void wait_async() {
  asm volatile("s_wait_asynccnt 0x0" ::: "memory");
}
__device__ __forceinline__ uint32_t lds_off32(const void* p) {
  return (uint32_t)(size_t)p;   // low 32 bits of generic LDS pointer = LDS offset
}

// CDNA5 Tensor Data Mover: DMA a 2D tile (tile0 x tile1 f32 elements, row
// stride stride0 elements) from global into LDS, inserting a 4-dword pad
// after every 128 dwords (pad_interval code 6, pad_amount code 3) so the
// LDS layout matches float[KCH][PHPADB]. Tracked by TENSORcnt.
__device__ __forceinline__ void tdm_load_2d(uint32_t lds_addr, const float* gaddr,
                                            uint32_t rem0, uint32_t rem1,
                                            uint32_t tile0, uint32_t tile1,
                                            uint32_t stride0) {
  uint64_t ga = (uint64_t)(size_t)gaddr;
  u32x4 g0;
  g0[0] = 1u;                                               // count=1, user D#
  g0[1] = lds_addr;                                         // lds_addr[31:0]
  g0[2] = (uint32_t)ga;                                     // global_addr[31:0]
  g0[3] = (uint32_t)((ga >> 32) & 0x01FFFFFFu) | (2u << 30);// addr[56:32], type=2
  u32x8 g1;
  g1[0] = (2u << 16) | (1u << 20) | (6u << 22) | (3u << 25);// 4B elems, pad 4dw/128dw
  g1[1] = (rem0 & 0xFFFFu) << 16;                           // tensor_dim0 lo16
  g1[2] = (rem0 >> 16) | ((rem1 & 0xFFFFu) << 16);          // dim0 hi / dim1 lo
  g1[3] = (rem1 >> 16) | (tile0 << 16);                     // dim1 hi / tile_dim0
  g1[4] = tile1;                                            // tile_dim1 (tile_dim2=0)
  g1[5] = stride0;                                          // dim0_stride lo32
  g1[6] = 0u;
  g1[7] = 0u;
  asm volatile("tensor_load_to_lds %0, %1" :: "s"(g0), "s"(g1) : "memory");
}

// ---------------------------------------------------------------- zeroing
__global__ void zero_out_kernel(float* __restrict__ out, int n4) {
  int i = blockIdx.x * blockDim.x + threadIdx.x;
  if (i < n4) ((float4*)out)[i] = make_float4(0.f, 0.f, 0.f, 0.f);
}

__global__ void zero_counts_kernel(int* __restrict__ counts) {
  if (threadIdx.x < EXP) counts[threadIdx.x] = 0;
}

// ---------------------------------------------------------------- routing
__global__ void routing_kernel(const int* __restrict__ sel,
                               const float* __restrict__ rw,
                               int* __restrict__ counts,
                               int* __restrict__ tok_list,
                               float* __restrict__ w_list) {
  int t = blockIdx.x * blockDim.x + threadIdx.x;
  if (t >= T_TOK) return;
  int e0 = sel[t * TOPK + 0];
  int e1 = sel[t * TOPK + 1];
  float w0 = rw[t * TOPK + 0];
  float w1 = rw[t * TOPK + 1];
  if (e0 == e1) { w0 += w1; e1 = -1; }
  {
    int idx = atomicAdd(&counts[e0], 1);
    tok_list[e0 * T_TOK + idx] = t;
    w_list[e0 * T_TOK + idx] = w0;
  }
  if (e1 >= 0) {
    int idx = atomicAdd(&counts[e1], 1);
    tok_list[e1 * T_TOK + idx] = t;
    w_list[e1 * T_TOK + idx] = w1;
  }
}

__global__ void scan_counts_kernel(const int* __restrict__ counts,
                                   int* __restrict__ offs) {
  if (threadIdx.x == 0) {
    int s = 0;
    for (int e = 0; e < EXP; ++e) { offs[e] = s; s += counts[e]; }
  }
}

// ================================================================ Phase A
// act[row, :] = gelu(x @ gate_w) * (x @ up_w), expert-grouped rows.
// 128x128 tile per block; weight tiles via TDM, gathered x tile via async.
__global__ __launch_bounds__(256)
void moe_gateup_kernel(const float* __restrict__ x,
                       const float* __restrict__ gate_w,
                       const float* __restrict__ up_w,
                       const int* __restrict__ counts,
                       const int* __restrict__ offs,
                       const int* __restrict__ tok_list,
                       float* __restrict__ act) {
  const int e    = blockIdx.z;
  const int tile = blockIdx.y;
  const int i0   = blockIdx.x * NT;
  const int cnt  = counts[e];
  if (tile * MT >= cnt) return;
  const int rowbase = tile * MT;
  const int arow0   = offs[e] + rowbase;

  __shared__ int toks[MT];
  __shared__ __attribute__((aligned(16))) float xs[2][MT][PHPADX];
  __shared__ __attribute__((aligned(16))) float bgs[2][KCH][PHPADB];
  __shared__ __attribute__((aligned(16))) float bus[2][KCH][PHPADB];

  const int tid = threadIdx.x;
  const int lane = tid & 31, wave = tid >> 5;
  const int wr = wave >> 2, wc = wave & 3;     // 2x4 wave grid: 64x32 per wave
  const int half = lane >> 4, l15 = lane & 15, ksel = half * 2;

  if (tid < MT) {
    int idx = rowbase + tid;
    toks[tid] = tok_list[e * T_TOK + (idx < cnt ? idx : cnt - 1)];
  }
  __syncthreads();

  const float* gw = gate_w + (size_t)e * HID * INTER;
  const float* uw = up_w   + (size_t)e * HID * INTER;

  auto stage = [&](int buf, int k0) {
#pragma unroll
    for (int t = 0; t < 4; ++t) {            // x tile: 128 gathered rows x 32 floats
      int c = tid + t * 256;                 // 1024 x 16B async chunks
      int row = c >> 3, seg = c & 7;
      async_copy_b128(lds_off32(&xs[buf][row][seg * 4]),
                      x + (size_t)toks[row] * HID + k0 + seg * 4);
    }
    if (wave == 0) {                          // gate tile: one TDM DMA
      tdm_load_2d(lds_off32(&bgs[buf][0][0]), gw + (size_t)k0 * INTER + i0,
                  INTER - i0, HID - k0, NT, KCH, INTER);
    } else if (wave == 1) {                   // up tile: one TDM DMA
      tdm_load_2d(lds_off32(&bus[buf][0][0]), uw + (size_t)k0 * INTER + i0,
                  INTER - i0, HID - k0, NT, KCH, INTER);
    }
  };

  const v8f vz = {};
  v8f gacc[4][2], uacc[4][2];
#pragma unroll
  for (int mt = 0; mt < 4; ++mt)
#pragma unroll
    for (int nt = 0; nt < 2; ++nt) { gacc[mt][nt] = vz; uacc[mt][nt] = vz; }

  stage(0, 0);
  wait_async();
  __builtin_amdgcn_s_wait_tensorcnt(0);
  __syncthreads();

  int buf = 0;
  for (int k0 = 0; k0 < HID; k0 += KCH) {
    if (k0 + KCH < HID) stage(buf ^ 1, k0 + KCH);   // prefetch next chunk

#pragma unroll
    for (int kk = 0; kk < KCH; kk += 4) {
      v2f a[4];
#pragma unroll
      for (int mt = 0; mt < 4; ++mt) {
        int m = wr * 64 + mt * 16 + l15;
        a[mt].x = xs[buf][m][kk + ksel];
        a[mt].y = xs[buf][m][kk + ksel + 1];
      }
#pragma unroll
      for (int nt = 0; nt < 2; ++nt) {
        int col = wc * 32 + nt * 16 + l15;
        v2f bg, bu;
        bg.x = bgs[buf][kk + ksel][col];
        bg.y = bgs[buf][kk + ksel + 1][col];
        bu.x = bus[buf][kk + ksel][col];
        bu.y = bus[buf][kk + ksel + 1][col];
#pragma unroll
        for (int mt = 0; mt < 4; ++mt) {
          gacc[mt][nt] = wmma_f32(a[mt], bg, gacc[mt][nt]);
          uacc[mt][nt] = wmma_f32(a[mt], bu, uacc[mt][nt]);
        }
      }
    }
    wait_async();
    __builtin_amdgcn_s_wait_tensorcnt(0);
    __syncthreads();
    buf ^= 1;
  }

#pragma unroll
  for (int mt = 0; mt < 4; ++mt)
#pragma unroll
    for (int nt = 0; nt < 2; ++nt)
#pragma unroll
      for (int r = 0; r < 8; ++r) {
        int m = wr * 64 + mt * 16 + r + half * 8;
        if (rowbase + m < cnt) {
          int col = i0 + wc * 32 + nt * 16 + l15;
          act[(size_t)(arow0 + m) * INTER + col] =
              gelu_tanh(gacc[mt][nt][r]) * uacc[mt][nt][r];
        }
      }
}

// ================================================================ Phase B
// out[tok, :] += w * (act @ down_w), atomic scatter over tokens.
__global__ __launch_bounds__(256)
void moe_down_kernel(const float* __restrict__ act,
                     const float* __restrict__ down_w,
                     const int* __restrict__ counts,
                     const int* __restrict__ offs,
                     const int* __restrict__ tok_list,
                     const float* __restrict__ w_list,
                     float* __restrict__ out) {
  const int e    = blockIdx.z;
  const int tile = blockIdx.y;
  const int h0   = blockIdx.x * NT;
  const int cnt  = counts[e];
  if (tile * MT >= cnt) return;
  const int rowbase = tile * MT;
  const int arow0   = offs[e];

  __shared__ int   toks[MT];
  __shared__ float wts[MT];
  __shared__ __attribute__((aligned(16))) float ats[2][MT][PHPADX];
  __shared__ __attribute__((aligned(16))) float bds[2][KCH][PHPADB];

  const int tid = threadIdx.x;
  const int lane = tid & 31, wave = tid >> 5;
  const int wr = wave >> 2, wc = wave & 3;
  const int half = lane >> 4, l15 = lane & 15, ksel = half * 2;

  if (tid < MT) {
    int idx = rowbase + tid;
    int cidx = idx < cnt ? idx : cnt - 1;
    toks[tid] = tok_list[e * T_TOK + cidx];
    wts[tid]  = idx < cnt ? w_list[e * T_TOK + cidx] : 0.0f;
  }
  __syncthreads();

  const float* dw = down_w + (size_t)e * INTER * HID;

  auto stage = [&](int buf, int k0) {
#pragma unroll
    for (int t = 0; t < 4; ++t) {            // act tile: 128 rows x 32 floats
      int c = tid + t * 256;
      int row = c >> 3, seg = c & 7;
      int g = rowbase + row;
      int arow = arow0 + (g < cnt ? g : cnt - 1);
      async_copy_b128(lds_off32(&ats[buf][row][seg * 4]),
                      act + (size_t)arow * INTER + k0 + seg * 4);
    }
    if (wave == 0) {                          // down tile: one TDM DMA
      tdm_load_2d(lds_off32(&bds[buf][0][0]), dw + (size_t)k0 * HID + h0,
                  HID - h0, INTER - k0, NT, KCH, HID);
    }
  };

  const v8f vz = {};
  v8f oacc[4][2];
#pragma unroll
  for (int mt = 0; mt < 4; ++mt)
#pragma unroll
    for (int nt = 0; nt < 2; ++nt) oacc[mt][nt] = vz;

  stage(0, 0);
  wait_async();
  __builtin_amdgcn_s_wait_tensorcnt(0);
  __syncthreads();

  int buf = 0;
  for (int k0 = 0; k0 < INTER; k0 += KCH) {
    if (k0 + KCH < INTER) stage(buf ^ 1, k0 + KCH);

#pragma unroll
    for (int kk = 0; kk < KCH; kk += 4) {
      v2f a[4];
#pragma unroll
      for (int mt = 0; mt < 4; ++mt) {
        int m = wr * 64 + mt * 16 + l15;
        a[mt].x = ats[buf][m][kk + ksel];
        a[mt].y = ats[buf][m][kk + ksel + 1];
      }
#pragma unroll
      for (int nt = 0; nt < 2; ++nt) {
        int col = wc * 32 + nt * 16 + l15;
        v2f b;
        b.x = bds[buf][kk + ksel][col];
        b.y = bds[buf][kk + ksel + 1][col];
#pragma unroll
        for (int mt = 0; mt < 4; ++mt)
          oacc[mt][nt] = wmma_f32(a[mt], b, oacc[mt][nt]);
      }
    }
    wait_async();
    __builtin_amdgcn_s_wait_tensorcnt(0);
    __syncthreads();
    buf ^= 1;
  }

#pragma unroll
  for (int mt = 0; mt < 4; ++mt)
#pragma unroll
    for (int nt = 0; nt < 2; ++nt)
#pragma unroll
      for (int r = 0; r < 8; ++r) {
        int m = wr * 64 + mt * 16 + r + half * 8;
        if (rowbase + m < cnt) {
          int col = h0 + wc * 32 + nt * 16 + l15;
          unsafeAtomicAdd(&out[(size_t)toks[m] * HID + col],
                          oacc[mt][nt][r] * wts[m]);
        }
      }
}

// ================================================================ fused fallback
// (small-workspace path; 16 tokens/block)
__global__ __launch_bounds__(THREADS)
void moe_ffn_kernel(const float* __restrict__ x,
                    const float* __restrict__ gate_w,
                    const float* __restrict__ up_w,
                    const float* __restrict__ down_w,
                    const int* __restrict__ counts,
                    const int* __restrict__ tok_list,
                    const float* __restrict__ w_list,
                    float* __restrict__ out) {
  const int e = blockIdx.y;
  const int tile = blockIdx.x;
  const int cnt = counts[e];
  if (tile * TB >= cnt) return;

  __shared__ int   toks[TB];
  __shared__ float wts[TB];
  __shared__ float xs[TB][PADX];
  __shared__ float as[TB][PADA];

  const int tid = threadIdx.x;
  const int lane = tid & 31;
  const int wave = tid >> 5;

  if (tid < TB) {
    int idx = tile * TB + tid;
    int cidx = idx < cnt ? idx : cnt - 1;
    toks[tid] = tok_list[e * T_TOK + cidx];
    wts[tid]  = idx < cnt ? w_list[e * T_TOK + cidx] : 0.0f;
  }
  __syncthreads();

  const int half = lane >> 4;
  const int l15  = lane & 15;
  const int ksel = half * 2;
  const int mrow = l15;

  const float* gw = gate_w + (size_t)e * HID * INTER;
  const float* uw = up_w   + (size_t)e * HID * INTER;
  const float* dw = down_w + (size_t)e * INTER * HID;

  const v8f vzero = {};
  v8f gacc[8], uacc[8];
#pragma unroll
  for (int i = 0; i < 8; ++i) { gacc[i] = vzero; uacc[i] = vzero; }

  const int i0 = wave * 128;

  for (int k0 = 0; k0 < HID; k0 += KC) {
    {
      int idx = tid * 4;
      int m = idx / KC, kk = idx % KC;
      const float4 v = *(const float4*)(x + (size_t)toks[m] * HID + k0 + kk);
      xs[m][kk + 0] = v.x; xs[m][kk + 1] = v.y;
      xs[m][kk + 2] = v.z; xs[m][kk + 3] = v.w;
    }
    __syncthreads();

#pragma unroll 4
    for (int kk = 0; kk < KC; kk += 4) {
      v2f a;
      a.x = xs[mrow][kk + ksel];
      a.y = xs[mrow][kk + ksel + 1];
      const int krow = k0 + kk + ksel;
#pragma unroll
      for (int nt = 0; nt < 8; ++nt) {
        const int col = i0 + nt * 16 + l15;
        v2f bg, bu;
        bg.x = gw[(size_t)krow * INTER + col];
        bg.y = gw[(size_t)(krow + 1) * INTER + col];
        bu.x = uw[(size_t)krow * INTER + col];
        bu.y = uw[(size_t)(krow + 1) * INTER + col];
        gacc[nt] = wmma_f32(a, bg, gacc[nt]);
        uacc[nt] = wmma_f32(a, bu, uacc[nt]);
      }
    }
    __syncthreads();
  }

#pragma unroll
  for (int nt = 0; nt < 8; ++nt)
#pragma unroll
    for (int r = 0; r < 8; ++r) {
      const int m = r + half * 8;
      const int col = i0 + nt * 16 + l15;
      as[m][col] = gelu_tanh(gacc[nt][r]) * uacc[nt][r];
    }
  __syncthreads();

  const int h0 = wave * 256;
  v8f oacc[16];
#pragma unroll
  for (int i = 0; i < 16; ++i) oacc[i] = vzero;

  for (int k = 0; k < INTER; k += 4) {
    v2f a;
    a.x = as[mrow][k + ksel];
    a.y = as[mrow][k + ksel + 1];
    const int krow = k + ksel;
#pragma unroll
    for (int ht = 0; ht < 16; ++ht) {
      const int col = h0 + ht * 16 + l15;
      v2f b;
      b.x = dw[(size_t)krow * HID + col];
      b.y = dw[(size_t)(krow + 1) * HID + col];
      oacc[ht] = wmma_f32(a, b, oacc[ht]);
    }
  }

#pragma unroll
  for (int ht = 0; ht < 16; ++ht)
#pragma unroll
    for (int r = 0; r < 8; ++r) {
      const int m = r + half * 8;
      const int col = h0 + ht * 16 + l15;
      unsafeAtomicAdd(&out[(size_t)toks[m] * HID + col], oacc[ht][r] * wts[m]);
    }
}

// ---------------------------------------------------------------- launch
extern "C" void kernel_launch(void* const* d_in, const int* in_sizes, int n_in,
                              void* d_out, int out_size, void* d_ws, size_t ws_size,
                              hipStream_t stream) {
  const float* x      = (const float*)d_in[0];
  const int*   sel    = (const int*)d_in[1];
  const float* rw     = (const float*)d_in[2];
  const float* gate_w = (const float*)d_in[3];
  const float* up_w   = (const float*)d_in[4];
  const float* down_w = (const float*)d_in[5];
  float* out = (float*)d_out;

  char* ws = (char*)d_ws;
  int*   counts   = (int*)(ws + OFF_COUNTS);
  int*   offs     = (int*)(ws + OFF_OFFS);
  int*   tok_list = (int*)(ws + OFF_TOK);
  float* w_list   = (float*)(ws + OFF_W);
  float* act      = (float*)(ws + OFF_ACT);

  const int n4 = (T_TOK * HID) / 4;
  zero_out_kernel<<<(n4 + 255) / 256, 256, 0, stream>>>(out, n4);
  zero_counts_kernel<<<1, 32, 0, stream>>>(counts);
  routing_kernel<<<(T_TOK + 255) / 256, 256, 0, stream>>>(
      sel, rw, counts, tok_list, w_list);
  scan_counts_kernel<<<1, 32, 0, stream>>>(counts, offs);

  if (ws_size >= WS_NEED) {
    // two-phase expert-grouped GEMMs (8x lower weight re-read than fused)
    dim3 ga(INTER / NT, T_TOK / MT, EXP);   // (8, 64, 8)
    moe_gateup_kernel<<<ga, 256, 0, stream>>>(x, gate_w, up_w, counts, offs,
                                              tok_list, act);
    dim3 gb(HID / NT, T_TOK / MT, EXP);     // (16, 64, 8)
    moe_down_kernel<<<gb, 256, 0, stream>>>(act, down_w, counts, offs,
                                            tok_list, w_list, out);
  } else {
    dim3 grid(T_TOK / TB, EXP);
    moe_ffn_kernel<<<grid, THREADS, 0, stream>>>(
        x, gate_w, up_w, down_w, counts, tok_list, w_list, out);
  }
}